// BipartiteGNN_71408126263375
// MI455X (gfx1250) — compile-verified
//
#include <hip/hip_runtime.h>

typedef float v2f __attribute__((ext_vector_type(2)));
typedef float v8f __attribute__((ext_vector_type(8)));

#define F_IN 128
#define HID  256
#define CLS  64

// ---------------------------------------------------------------------------
// Degree counts for both edge directions (p_idx destinations and u_idx dsts).
// ---------------------------------------------------------------------------
__global__ void count_deg_kernel(const int* __restrict__ edge_index,
                                 float* __restrict__ cnt_p,
                                 float* __restrict__ cnt_u, int E) {
    int e = blockIdx.x * blockDim.x + threadIdx.x;
    if (e < E) {
        int u = edge_index[e];       // row 0: u_idx
        int p = edge_index[E + e];   // row 1: p_idx
        unsafeAtomicAdd(&cnt_u[u], 1.0f);
        unsafeAtomicAdd(&cnt_p[p], 1.0f);
    }
}

// ---------------------------------------------------------------------------
// Edge-parallel scatter add: agg[edst[e]][:] += xsrc[esrc[e]][:]
// One thread handles a float4 chunk of one edge; consecutive lanes cover
// consecutive feature chunks so the atomic stream per edge is coalesced.
// ---------------------------------------------------------------------------
template <int F>
__global__ void scatter_add_kernel(const float* __restrict__ xsrc,
                                   const int* __restrict__ esrc,
                                   const int* __restrict__ edst,
                                   float* __restrict__ agg, int E) {
    const int chunks = F / 4;
    long idx = (long)blockIdx.x * blockDim.x + threadIdx.x;
    long total = (long)E * chunks;
    if (idx >= total) return;
    int e = (int)(idx / chunks);
    int c = (int)(idx % chunks);
    int s = esrc[e];
    int d = edst[e];
    const float4 v = ((const float4*)(xsrc + (size_t)s * F))[c];
    float* dst = agg + (size_t)d * F + (size_t)c * 4;
    unsafeAtomicAdd(dst + 0, v.x);
    unsafeAtomicAdd(dst + 1, v.y);
    unsafeAtomicAdd(dst + 2, v.z);
    unsafeAtomicAdd(dst + 3, v.w);
}

// ---------------------------------------------------------------------------
// Fused SAGE update via CDNA5 WMMA (fp32, 16x16x4):
//   out = (agg/max(cnt,1)) @ Wl^T + bias + X @ Wr^T   (optional ReLU)
// The mean normalization is folded into the A-fragment load (scale per row).
//
// agg, X: [N, K] row-major.  Wl, Wr: [OUT, K] row-major.  out: [N, OUT].
// One wave owns CT column tiles of a 16-row stripe: 16 x (16*CT) of C.
// A/X fragments are loaded once per k-step and reused across all CT tiles
// (2*CT WMMAs per k-step), cutting activation traffic CT-fold.
// Block = 4 waves covering 16 rows x (64*CT) cols.
//
// Fragment layouts per ISA 7.12.2:
//   A 16x4 : lane L (0-15) row M=L, VGPR0/1 = K0/K1; lanes 16-31 = K2/K3.
//   B 4x16 : same pattern with N = lane%16 (B loaded K-major from W rows).
//   C/D    : VGPR r, lane l -> M = r + 8*(l/16), N = l%16.
// N is a multiple of 16 (50000 = 3125*16) so EXEC is all-ones at WMMA.
// ---------------------------------------------------------------------------
template <int K, int OUT, int CT, bool RELU>
__launch_bounds__(128)
__global__ void sage_gemm_wmma_kernel(const float* __restrict__ Agg,
                                      const float* __restrict__ cnt,
                                      const float* __restrict__ X,
                                      const float* __restrict__ Wl,
                                      const float* __restrict__ Wr,
                                      const float* __restrict__ bias,
                                      float* __restrict__ out) {
    const int lane = threadIdx.x & 31;
    const int wave = threadIdx.x >> 5;
    const int row0 = blockIdx.x * 16;
    const int col0 = blockIdx.y * (64 * CT) + wave * (16 * CT);
    const int m    = lane & 15;           // row (A/X) or col (W) within tile
    const int koff = (lane >> 4) * 2;     // K sub-offset per half-wave

    const float* arow = Agg + (size_t)(row0 + m) * K;
    const float* xrow = X   + (size_t)(row0 + m) * K;
    const float* wlp[CT];
    const float* wrp[CT];
    #pragma unroll
    for (int t = 0; t < CT; ++t) {
        wlp[t] = Wl + (size_t)(col0 + 16 * t + m) * K;
        wrp[t] = Wr + (size_t)(col0 + 16 * t + m) * K;
    }

    // Mean normalization folded into the A operand.
    const float scale = 1.0f / fmaxf(cnt[row0 + m], 1.0f);

    v8f acc[CT];
    #pragma unroll
    for (int t = 0; t < CT; ++t)
        acc[t] = (v8f){0.f, 0.f, 0.f, 0.f, 0.f, 0.f, 0.f, 0.f};

    #pragma unroll 2
    for (int k = 0; k < K; k += 4) {
        v2f af, df;
        af.x = arow[k + koff] * scale;
        af.y = arow[k + koff + 1] * scale;
        df.x = xrow[k + koff];
        df.y = xrow[k + koff + 1];
        #pragma unroll
        for (int t = 0; t < CT; ++t) {
            v2f bf, gf;
            bf.x = wlp[t][k + koff];
            bf.y = wlp[t][k + koff + 1];
            gf.x = wrp[t][k + koff];
            gf.y = wrp[t][k + koff + 1];
            // acc += mean_frag * Wl_frag   (aggregated-neighbor path)
            acc[t] = __builtin_amdgcn_wmma_f32_16x16x4_f32(
                         false, af, false, bf, (short)0, acc[t], false, false);
            // acc += X_frag * Wr_frag      (root path)
            acc[t] = __builtin_amdgcn_wmma_f32_16x16x4_f32(
                         false, df, false, gf, (short)0, acc[t], false, false);
        }
    }

    const int mbase = (lane >> 4) * 8;
    #pragma unroll
    for (int t = 0; t < CT; ++t) {
        const int col = col0 + 16 * t + m;
        const float b = bias[col];
        #pragma unroll
        for (int r = 0; r < 8; ++r) {
            float v = acc[t][r] + b;
            if (RELU) v = fmaxf(v, 0.f);
            out[(size_t)(row0 + mbase + r) * OUT + col] = v;
        }
    }
}

// ---------------------------------------------------------------------------
// Host-side orchestration.
// ---------------------------------------------------------------------------
extern "C" void kernel_launch(void* const* d_in, const int* in_sizes, int n_in,
                              void* d_out, int out_size, void* d_ws, size_t ws_size,
                              hipStream_t stream) {
    const float* x_user    = (const float*)d_in[0];
    const float* x_product = (const float*)d_in[1];
    const int*   edge      = (const int*)d_in[2];
    const float* w_u1_l = (const float*)d_in[3];
    const float* b_u1   = (const float*)d_in[4];
    const float* w_u1_r = (const float*)d_in[5];
    const float* w_p1_l = (const float*)d_in[6];
    const float* b_p1   = (const float*)d_in[7];
    const float* w_p1_r = (const float*)d_in[8];
    const float* w_u2_l = (const float*)d_in[9];
    const float* b_u2   = (const float*)d_in[10];
    const float* w_u2_r = (const float*)d_in[11];
    const float* w_p2_l = (const float*)d_in[12];
    const float* b_p2   = (const float*)d_in[13];
    const float* w_p2_r = (const float*)d_in[14];

    const int N = in_sizes[0] / F_IN;     // 50000
    const int E = in_sizes[2] / 2;        // 800000
    const int* u_idx = edge;              // row 0
    const int* p_idx = edge + E;          // row 1

    // Workspace layout (floats)
    float* ws = (float*)d_ws;
    float* cnt_p = ws;                              // [N]
    float* cnt_u = cnt_p + N;                       // [N]
    float* agg_a = cnt_u + N;                       // [N,128] seg(x_user[u],p)
    float* agg_b = agg_a + (size_t)N * F_IN;        // [N,128] seg(x_prod[p],u)
    float* agg_c = agg_b + (size_t)N * F_IN;        // [N,256] seg(h_u[u],p)
    float* agg_d = agg_c + (size_t)N * HID;         // [N,256] seg(h_p[p],u)
    float* h_u   = agg_d + (size_t)N * HID;         // [N,256] x_user_1
    float* h_p   = h_u   + (size_t)N * HID;         // [N,256] x_product_1

    // Zero the atomic-accumulated regions (cnt + 4 agg buffers, contiguous).
    size_t zero_floats = (size_t)N * (2 + 2 * F_IN + 2 * HID);
    hipMemsetAsync(d_ws, 0, zero_floats * sizeof(float), stream);

    const int TB = 256;

    // Degree counts.
    count_deg_kernel<<<(E + TB - 1) / TB, TB, 0, stream>>>(edge, cnt_p, cnt_u, E);

    // Layer-1 aggregation.
    {
        long tot = (long)E * (F_IN / 4);
        int blocks = (int)((tot + TB - 1) / TB);
        scatter_add_kernel<F_IN><<<blocks, TB, 0, stream>>>(x_user,    u_idx, p_idx, agg_a, E);
        scatter_add_kernel<F_IN><<<blocks, TB, 0, stream>>>(x_product, p_idx, u_idx, agg_b, E);
    }

    // Layer-1 fused (mean-normalize + GEMMs + bias + ReLU) -> hidden features.
    {
        dim3 grid(N / 16, 1);   // CT=4: one block row covers all 256 cols
        sage_gemm_wmma_kernel<F_IN, HID, 4, true><<<grid, 128, 0, stream>>>(
            agg_a, cnt_p, x_product, w_u1_l, w_u1_r, b_u1, h_u);
        sage_gemm_wmma_kernel<F_IN, HID, 4, true><<<grid, 128, 0, stream>>>(
            agg_b, cnt_u, x_user, w_p1_l, w_p1_r, b_p1, h_p);
    }

    // Layer-2 aggregation on hidden features.
    {
        long tot = (long)E * (HID / 4);
        int blocks = (int)((tot + TB - 1) / TB);
        scatter_add_kernel<HID><<<blocks, TB, 0, stream>>>(h_u, u_idx, p_idx, agg_c, E);
        scatter_add_kernel<HID><<<blocks, TB, 0, stream>>>(h_p, p_idx, u_idx, agg_d, E);
    }

    // Layer-2 fused GEMMs (no ReLU) -> outputs, concatenated [user | product].
    float* out_user = (float*)d_out;
    float* out_prod = out_user + (size_t)N * CLS;
    {
        dim3 grid(N / 16, 1);   // CT=1: 4 waves cover the 64 cols
        sage_gemm_wmma_kernel<HID, CLS, 1, false><<<grid, 128, 0, stream>>>(
            agg_c, cnt_p, h_p, w_u2_l, w_u2_r, b_u2, out_user);
        sage_gemm_wmma_kernel<HID, CLS, 1, false><<<grid, 128, 0, stream>>>(
            agg_d, cnt_u, h_u, w_p2_l, w_p2_r, b_p2, out_prod);
    }
}